// MyLoss_82686710383027
// MI455X (gfx1250) — compile-verified
//
#include <hip/hip_runtime.h>

// Shapes from the reference: (8,16,6,256,256) -> 768 rows of 65536 fp32 each.
#define HW      65536
#define THREADS 256
#define WAVES   (THREADS / 32)

typedef __attribute__((ext_vector_type(4))) float v4f;
typedef __attribute__((ext_vector_type(2))) float v2f;
typedef __attribute__((ext_vector_type(8))) float v8f;

// Cross-lane (wave32) sum using the matrix pipe:
// A = (s_lane, 0) per lane, B = all-ones  =>  D[m][n] = s_m + s_{m+16}.
// Lanes 0-15 hold rows 0-7 in their 8 C VGPRs, lanes 16-31 hold rows 8-15,
// so (sum of my 8 VGPRs) + shfl_xor(16) = full 32-lane sum in every lane.
__device__ __forceinline__ float wave_sum_wmma(float s) {
    v2f a; a.x = s;    a.y = 0.0f;
    v2f b; b.x = 1.0f; b.y = 1.0f;
    v8f c = {};
    c = __builtin_amdgcn_wmma_f32_16x16x4_f32(
        /*neg_a=*/false, a, /*neg_b=*/false, b,
        /*c_mod=*/(short)0, c, /*reuse_a=*/false, /*reuse_b=*/false);
    float t = ((c[0] + c[1]) + (c[2] + c[3])) + ((c[4] + c[5]) + (c[6] + c[7]));
    t += __shfl_xor(t, 16, 32);
    return t;
}

__global__ void __launch_bounds__(THREADS)
row_corr_kernel(const float* __restrict__ x,
                const float* __restrict__ y,
                float* __restrict__ row_r) {
    const int  row  = blockIdx.x;
    const int  tid  = threadIdx.x;
    const long long base = (long long)row * HW;

    const v4f* px = (const v4f*)(x + base) + tid;
    const v4f* py = (const v4f*)(y + base) + tid;

    v4f sx = {}, sy = {}, sx2 = {}, sy2 = {}, sxy = {};

    const int iters = HW / (THREADS * 4);   // 64
#pragma unroll 4
    for (int i = 0; i < iters; ++i) {
        v4f xv = __builtin_nontemporal_load(px);   // stream: each byte read once
        v4f yv = __builtin_nontemporal_load(py);
        px += THREADS;
        py += THREADS;
        sx  += xv;
        sy  += yv;
        sx2 += xv * xv;     // v_fmac_f32
        sy2 += yv * yv;
        sxy += xv * yv;
    }

    // Horizontal add of the per-lane float4 accumulators.
    float S[5];
    S[0] = (sx.x  + sx.y)  + (sx.z  + sx.w);
    S[1] = (sy.x  + sy.y)  + (sy.z  + sy.w);
    S[2] = (sx2.x + sx2.y) + (sx2.z + sx2.w);
    S[3] = (sy2.x + sy2.y) + (sy2.z + sy2.w);
    S[4] = (sxy.x + sxy.y) + (sxy.z + sxy.w);

    // Wave-level reduction on the matrix pipe (EXEC all-1s here: no divergence).
#pragma unroll
    for (int q = 0; q < 5; ++q)
        S[q] = wave_sum_wmma(S[q]);

    __shared__ float red[WAVES][5];
    const int wave = tid >> 5;
    const int lane = tid & 31;
    if (lane == 0) {
#pragma unroll
        for (int q = 0; q < 5; ++q) red[wave][q] = S[q];
    }
    __syncthreads();

    if (tid == 0) {
        float T[5] = {0.f, 0.f, 0.f, 0.f, 0.f};
        for (int w = 0; w < WAVES; ++w)
#pragma unroll
            for (int q = 0; q < 5; ++q) T[q] += red[w][q];

        const float N      = (float)HW;
        const float invN   = 1.0f / N;
        const float invNm1 = 1.0f / (N - 1.0f);

        const float Sx = T[0], Sy = T[1], Sx2 = T[2], Sy2 = T[3], Sxy = T[4];
        const float vx  = (Sx2 - Sx * Sx * invN) * invNm1;
        const float vy  = (Sy2 - Sy * Sy * invN) * invNm1;
        const float Ss  = Sx + Sy;
        const float vxy = (Sx2 + 2.0f * Sxy + Sy2 - Ss * Ss * invN) * invNm1;

        const float r = 1.0f - 0.5f * (vxy - vx - vy) / sqrtf(fabsf(vx * vy));
        row_r[row] = r;
    }
}

__global__ void __launch_bounds__(256)
mean_kernel(const float* __restrict__ row_r, float* __restrict__ out, int rows) {
    __shared__ float buf[256];
    float s = 0.0f;
    for (int i = threadIdx.x; i < rows; i += 256)
        s += row_r[i];
    buf[threadIdx.x] = s;
    __syncthreads();
    for (int off = 128; off > 0; off >>= 1) {
        if (threadIdx.x < (unsigned)off) buf[threadIdx.x] += buf[threadIdx.x + off];
        __syncthreads();
    }
    if (threadIdx.x == 0) out[0] = buf[0] / (float)rows;
}

extern "C" void kernel_launch(void* const* d_in, const int* in_sizes, int n_in,
                              void* d_out, int out_size, void* d_ws, size_t ws_size,
                              hipStream_t stream) {
    const float* x   = (const float*)d_in[0];
    const float* y   = (const float*)d_in[1];
    float*       out = (float*)d_out;
    float*       ws  = (float*)d_ws;   // per-row r values (rows * 4 bytes)

    const int rows = in_sizes[0] / HW;  // 768 for the reference shape

    row_corr_kernel<<<rows, THREADS, 0, stream>>>(x, y, ws);
    mean_kernel<<<1, 256, 0, stream>>>(ws, out, rows);
}